// GATModel_45389214384864
// MI455X (gfx1250) — compile-verified
//
#include <hip/hip_runtime.h>

// ---------------------------------------------------------------------------
// Two-layer GAT (PyG GATConv math) for gfx1250.
//   L1: h1 = x @ W1^T            [N,64], 4 heads x 16 ch   (WMMA f32 16x16x4)
//       edge softmax over dst, weighted aggregate, +b1, relu
//   L2: h2 = out1 @ W2^T         [N,32], 1 head x 32 ch    (WMMA f32 16x16x4)
//       edge softmax, aggregate, +b2  -> d_out [N,32] f32
// ---------------------------------------------------------------------------

#define GAT_NEG_SLOPE 0.2f

typedef __attribute__((ext_vector_type(2))) float v2f;
typedef __attribute__((ext_vector_type(8))) float v8f;

// order-preserving float -> uint key (for atomic max)
__device__ __forceinline__ unsigned fkey(float f) {
  unsigned u = __float_as_uint(f);
  return (u & 0x80000000u) ? ~u : (u | 0x80000000u);
}
__device__ __forceinline__ float funkey(unsigned k) {
  return (k & 0x80000000u) ? __uint_as_float(k & 0x7fffffffu)
                           : __uint_as_float(~k);
}
__device__ __forceinline__ float leaky(float v) {
  return v > 0.f ? v : GAT_NEG_SLOPE * v;
}

__global__ void zero_kernel(float* __restrict__ p, int n) {
  int i = blockIdx.x * blockDim.x + threadIdx.x;
  if (i < n) p[i] = 0.f;
}

// H[M x Nc] = A[M x K] * W[Nc x K]^T  via V_WMMA_F32_16X16X4_F32
// One wave computes one 16x16 tile. M, Nc multiples of 16; K multiple of 4.
__global__ void gemm_wmma_f32(const float* __restrict__ A,
                              const float* __restrict__ W,
                              float* __restrict__ H,
                              int M, int K, int Nc) {
  int wave = (blockIdx.x * blockDim.x + threadIdx.x) >> 5;
  int lane = threadIdx.x & 31;
  int ntiles = Nc >> 4;
  int total = (M >> 4) * ntiles;
  if (wave >= total) return;                 // wave-uniform: EXEC stays full
  int mbase = (wave / ntiles) << 4;
  int nbase = (wave % ntiles) << 4;
  int half = lane >> 4;                      // 0: lanes 0-15, 1: lanes 16-31
  int l = lane & 15;
  const float* arow = A + (size_t)(mbase + l) * K;   // A row for this lane
  const float* wrow = W + (size_t)(nbase + l) * K;   // B col j -> W row j
  v8f acc = {};
  for (int kb = 0; kb < K; kb += 4) {
    int ka = kb + 2 * half;                  // lanes 0-15: K=kb,kb+1 ; 16-31: kb+2,kb+3
    v2f a = *(const v2f*)(arow + ka);
    v2f b = *(const v2f*)(wrow + ka);
    acc = __builtin_amdgcn_wmma_f32_16x16x4_f32(
        /*neg_a=*/false, a, /*neg_b=*/false, b,
        /*c_mod=*/(short)0, acc, /*reuse_a=*/false, /*reuse_b=*/false);
  }
  // C/D layout: VGPR r -> M = mbase + r + 8*half, N = nbase + (lane&15)
  float* hp = H + (size_t)(mbase + 8 * half) * Nc + nbase + l;
  #pragma unroll
  for (int r = 0; r < 8; ++r) hp[(size_t)r * Nc] = acc[r];
}

// alpha_src[n,h] = sum_c h[n,h,c]*a_src[h,c]   (and same for a_dst)
__global__ void alpha_kernel(const float* __restrict__ h,
                             const float* __restrict__ a_src,
                             const float* __restrict__ a_dst,
                             float* __restrict__ as, float* __restrict__ ad,
                             int n, int heads, int C) {
  int tid = blockIdx.x * blockDim.x + threadIdx.x;
  if (tid >= n * heads) return;
  int node = tid / heads, hd = tid % heads;
  const float* hp = h + (size_t)node * heads * C + hd * C;
  const float* s = a_src + hd * C;
  const float* d = a_dst + hd * C;
  float ss = 0.f, dd = 0.f;
  for (int c = 0; c < C; ++c) { float v = hp[c]; ss += v * s[c]; dd += v * d[c]; }
  as[tid] = ss;
  ad[tid] = dd;
}

// pass 1: segment max of leaky(as[src]+ad[dst]) over dst (via uint key atomicMax)
__global__ void edge_max_kernel(const int* __restrict__ src, const int* __restrict__ dst,
                                const float* __restrict__ as, const float* __restrict__ ad,
                                unsigned* __restrict__ key, int Ecnt, int Ncnt, int heads) {
  int e = blockIdx.x * blockDim.x + threadIdx.x;
  if (e >= Ecnt + Ncnt) return;
  int s, d;
  if (e < Ecnt) { s = src[e]; d = dst[e]; } else { s = d = e - Ecnt; }
  for (int hd = 0; hd < heads; ++hd) {
    float v = leaky(as[s * heads + hd] + ad[d * heads + hd]);
    atomicMax(&key[d * heads + hd], fkey(v));
  }
}

// pass 2: denom[dst] += exp(e - m[dst])
__global__ void edge_den_kernel(const int* __restrict__ src, const int* __restrict__ dst,
                                const float* __restrict__ as, const float* __restrict__ ad,
                                const unsigned* __restrict__ key, float* __restrict__ den,
                                int Ecnt, int Ncnt, int heads) {
  int e = blockIdx.x * blockDim.x + threadIdx.x;
  if (e >= Ecnt + Ncnt) return;
  int s, d;
  if (e < Ecnt) { s = src[e]; d = dst[e]; } else { s = d = e - Ecnt; }
  for (int hd = 0; hd < heads; ++hd) {
    float v = leaky(as[s * heads + hd] + ad[d * heads + hd]);
    float m = funkey(key[d * heads + hd]);
    atomicAdd(&den[d * heads + hd], __expf(v - m));
  }
}

// pass 3: out[dst] += h[src] * softmax_weight.  CT consecutive lanes per edge
// -> coalesced h reads and out atomics.
template <int HEADS_T, int CT>
__global__ void edge_agg_kernel(const int* __restrict__ src, const int* __restrict__ dst,
                                const float* __restrict__ as, const float* __restrict__ ad,
                                const unsigned* __restrict__ key, const float* __restrict__ den,
                                const float* __restrict__ h, float* __restrict__ out,
                                int Ecnt, int Ncnt) {
  int g = blockIdx.x * blockDim.x + threadIdx.x;
  int e = g / CT;
  int c = g % CT;
  if (e >= Ecnt + Ncnt) return;
  int s, d;
  if (e < Ecnt) { s = src[e]; d = dst[e]; } else { s = d = e - Ecnt; }
  #pragma unroll
  for (int hd = 0; hd < HEADS_T; ++hd) {
    float v = leaky(as[s * HEADS_T + hd] + ad[d * HEADS_T + hd]);
    float m = funkey(key[d * HEADS_T + hd]);
    float w = __expf(v - m) / den[d * HEADS_T + hd];
    atomicAdd(&out[((size_t)d * HEADS_T + hd) * CT + c],
              h[((size_t)s * HEADS_T + hd) * CT + c] * w);
  }
}

__global__ void bias_act_kernel(float* __restrict__ out, const float* __restrict__ b,
                                int total, int stride, int do_relu) {
  int i = blockIdx.x * blockDim.x + threadIdx.x;
  if (i >= total) return;
  float v = out[i] + b[i % stride];
  out[i] = do_relu ? fmaxf(v, 0.f) : v;
}

static inline int cdiv(long a, long b) { return (int)((a + b - 1) / b); }

extern "C" void kernel_launch(void* const* d_in, const int* in_sizes, int n_in,
                              void* d_out, int out_size, void* d_ws, size_t ws_size,
                              hipStream_t stream) {
  const int IN = 64, HID = 64, HEADS = 4, C1 = 16, OUT = 32;
  const int N = in_sizes[0] / IN;      // 50000
  const int E = in_sizes[1] / 2;       // 800000
  const int EP = E + N;                // edges + self loops

  const float* x      = (const float*)d_in[0];
  const int*   eidx   = (const int*)d_in[1];   // JAX x64 disabled -> int32
  const int*   src    = eidx;
  const int*   dst    = eidx + E;
  const float* W1     = (const float*)d_in[2];
  const float* a_src1 = (const float*)d_in[3];
  const float* a_dst1 = (const float*)d_in[4];
  const float* b1     = (const float*)d_in[5];
  const float* W2     = (const float*)d_in[6];
  const float* a_src2 = (const float*)d_in[7];
  const float* a_dst2 = (const float*)d_in[8];
  const float* b2     = (const float*)d_in[9];
  float* dout = (float*)d_out;

  // ---- workspace carve-out (floats) -------------------------------------
  float* ws = (float*)d_ws;
  size_t o = 0;
  float*    h1   = ws + o; o += (size_t)N * HID;
  float*    out1 = ws + o; o += (size_t)N * HID;
  float*    as1  = ws + o; o += (size_t)N * HEADS;
  float*    ad1  = ws + o; o += (size_t)N * HEADS;
  unsigned* key1 = (unsigned*)(ws + o); o += (size_t)N * HEADS;
  float*    den1 = ws + o; o += (size_t)N * HEADS;
  float*    h2   = ws + o; o += (size_t)N * OUT;
  float*    as2  = ws + o; o += (size_t)N;
  float*    ad2  = ws + o; o += (size_t)N;
  unsigned* key2 = (unsigned*)(ws + o); o += (size_t)N;
  float*    den2 = ws + o; o += (size_t)N;
  (void)ws_size; (void)n_in; (void)out_size;

  const int B = 256;

  // ---- init accumulators / softmax state --------------------------------
  zero_kernel<<<cdiv((long)N * HID, B), B, 0, stream>>>(out1, N * HID);
  zero_kernel<<<cdiv((long)N * OUT, B), B, 0, stream>>>(dout, N * OUT);
  zero_kernel<<<cdiv((long)N * HEADS, B), B, 0, stream>>>((float*)key1, N * HEADS);
  zero_kernel<<<cdiv((long)N * HEADS, B), B, 0, stream>>>(den1, N * HEADS);
  zero_kernel<<<cdiv((long)N, B), B, 0, stream>>>((float*)key2, N);
  zero_kernel<<<cdiv((long)N, B), B, 0, stream>>>(den2, N);

  // ---- layer 1 -----------------------------------------------------------
  {
    long waves = (long)(N / 16) * (HID / 16);
    gemm_wmma_f32<<<cdiv(waves * 32, B), B, 0, stream>>>(x, W1, h1, N, IN, HID);
  }
  alpha_kernel<<<cdiv((long)N * HEADS, B), B, 0, stream>>>(h1, a_src1, a_dst1,
                                                           as1, ad1, N, HEADS, C1);
  edge_max_kernel<<<cdiv(EP, B), B, 0, stream>>>(src, dst, as1, ad1, key1, E, N, HEADS);
  edge_den_kernel<<<cdiv(EP, B), B, 0, stream>>>(src, dst, as1, ad1, key1, den1, E, N, HEADS);
  edge_agg_kernel<HEADS, C1><<<cdiv((long)EP * C1, B), B, 0, stream>>>(
      src, dst, as1, ad1, key1, den1, h1, out1, E, N);
  bias_act_kernel<<<cdiv((long)N * HID, B), B, 0, stream>>>(out1, b1, N * HID, HID, 1);

  // ---- layer 2 -----------------------------------------------------------
  {
    long waves = (long)(N / 16) * (OUT / 16);
    gemm_wmma_f32<<<cdiv(waves * 32, B), B, 0, stream>>>(out1, W2, h2, N, HID, OUT);
  }
  alpha_kernel<<<cdiv((long)N, B), B, 0, stream>>>(h2, a_src2, a_dst2, as2, ad2, N, 1, OUT);
  edge_max_kernel<<<cdiv(EP, B), B, 0, stream>>>(src, dst, as2, ad2, key2, E, N, 1);
  edge_den_kernel<<<cdiv(EP, B), B, 0, stream>>>(src, dst, as2, ad2, key2, den2, E, N, 1);
  edge_agg_kernel<1, OUT><<<cdiv((long)EP * OUT, B), B, 0, stream>>>(
      src, dst, as2, ad2, key2, den2, h2, dout, E, N);
  bias_act_kernel<<<cdiv((long)N * OUT, B), B, 0, stream>>>(dout, b2, N * OUT, OUT, 0);
}